// VolumeConvolutionF_88467736363263
// MI455X (gfx1250) — compile-verified
//
#include <hip/hip_runtime.h>

// =====================================================================
// Circular 3D cross-correlation via DFT-as-WMMA-GEMM on gfx1250.
//   out[b,t] = sum_r v1[b,(r+t) mod V] * v2[b,r]
//   = IFFT3( FFT3(v1) * conj(FFT3(v2)) )
// Each 1D DFT-128 pass is a complex GEMM  C = W(128x128) * X(128x16384)
// executed with V_WMMA_F32_16X16X4_F32. Passes transform the contiguous
// axis and write transposed, cycling [z][y][x]->[x'][z][y]->[y'][x'][z]
// ->[z'][y'][x'], so all global accesses stay coalesced.
//
// v4: - LDS tiles stored line-major with XOR swizzle (conflict-free,
//       16B-contiguous); staging uses GLOBAL_LOAD_ASYNC_TO_LDS_B128
//       (ASYNCcnt path) with the exact builtin prototype (v4i AS1 src,
//       v4i AS3 dst, imm offset, imm cpol); float4 stores otherwise.
//     - Epilogue arithmetic selected at compile time (no dead +0 adds).
// =====================================================================

typedef float v2f __attribute__((ext_vector_type(2)));
typedef float v8f __attribute__((ext_vector_type(8)));
typedef int   v4i __attribute__((vector_size(16)));   // matches builtin proto

#define VDIM   128
#define NLINE  (VDIM * VDIM)          // 16384 lines per pass
#define VOL    (VDIM * VDIM * VDIM)   // 2097152 elements per volume
#define TILES  4                      // line-tiles (of 16) per workgroup
#define LINES_PER_WG (16 * TILES)     // 64

#if defined(__has_builtin)
#if __has_builtin(__builtin_amdgcn_global_load_async_to_lds_b128)
#define ASYNC_STAGE 1
#endif
#endif

// ---------------------------------------------------------------------
// One 16x16x4 f32 WMMA step: D = A(16x4) * B(4x16) + C
// A frag: lanes 0-15 -> rows M=0..15 with K={0,1}; lanes 16-31 -> K={2,3}
// B frag: lanes 0-15 -> cols N=0..15 with K={0,1}; lanes 16-31 -> K={2,3}
// C/D:    VGPR v, lanes 0-15 -> M=v; lanes 16-31 -> M=v+8; N=lane&15
// ---------------------------------------------------------------------
__device__ __forceinline__ v8f wmma_f32(v2f a, v2f b, v8f c) {
    return __builtin_amdgcn_wmma_f32_16x16x4_f32(
        /*neg_a=*/false, a, /*neg_b=*/false, b,
        /*c_mod=*/(short)0, c, /*reuse_a=*/false, /*reuse_b=*/false);
}

// LDS swizzle: line-major, physical = l*128 + (n ^ ((l&15)<<2)).
// - 16B chunks (n = 0 mod 4) stay contiguous (swizzle hits bits 2..5).
// - B-frag read (line = tile*16+col, samples kk+h, kk+h+1): bank
//   ((kk+h) ^ 4*col) % 64 -> 16 distinct banks per lane half; halves
//   differ mod 4 -> disjoint. Conflict-free ds_load_b64.
__device__ __forceinline__ int lds_addr(int l, int n) {
    return l * VDIM + (n ^ ((l & 15) << 2));
}

// 16-byte global->LDS copy: async DMA if the toolchain exposes it.
__device__ __forceinline__ void stage16(float* lds_dst, const float4* g_src) {
#ifdef ASYNC_STAGE
    typedef __attribute__((address_space(1))) v4i g4_t;   // __device__ v4i*
    typedef __attribute__((address_space(3))) v4i l4_t;   // __shared__ v4i*
    __builtin_amdgcn_global_load_async_to_lds_b128(
        (g4_t*)(uintptr_t)g_src,
        (l4_t*)(unsigned int)(uintptr_t)lds_dst,
        /*offset=*/0, /*cpol=*/0);
#else
    *(float4*)lds_dst = *g_src;
#endif
}

__device__ __forceinline__ void stage_fence() {
#ifdef ASYNC_STAGE
    asm volatile("s_wait_asynccnt 0" ::: "memory");
#endif
}

// ---------------------------------------------------------------------
// DFT twiddle matrices (row-major 128x128).
// Forward:  W = exp(-i*2*pi*k*n/128)
// Inverse:  W = exp(+i*2*pi*k*n/128) / 128   (1/128^3 over 3 passes)
// ---------------------------------------------------------------------
__global__ void twiddle_init_kernel(float* __restrict__ wf_r,
                                    float* __restrict__ wf_i,
                                    float* __restrict__ wv_r,
                                    float* __restrict__ wv_i) {
    int t = blockIdx.x * blockDim.x + threadIdx.x;   // 0..16383
    int k = t >> 7;
    int n = t & 127;
    int m = (k * n) & 127;                            // exact modular phase
    const float PI_OVER_64 = 0.049087385212340517f;   // pi/64
    float th = (float)m * PI_OVER_64;
    float c = cosf(th);
    float s = sinf(th);
    wf_r[t] = c;
    wf_i[t] = -s;
    wv_r[t] = c * (1.0f / 128.0f);
    wv_i[t] = s * (1.0f / 128.0f);
}

// ---------------------------------------------------------------------
// One DFT pass over the contiguous axis, transposed write-back.
// Input logical layout [a][b][c], c contiguous; line L = a*128+b is a
// contiguous 128-float run. Output written as out[k*16384 + L]
// (layout [c'][a][b], b contiguous for the next pass).
// Block = 256 threads (8 waves). Block handles 64 consecutive lines
// (4 tiles of 16); wave w computes k-tile [16w, 16w+16) for all 4 tiles.
// ---------------------------------------------------------------------
template <bool HAS_IMAG, bool WRITE_IMAG>
__global__ __launch_bounds__(256)
void dft_pass_kernel(const float* __restrict__ inR,
                     const float* __restrict__ inI,
                     const float* __restrict__ Wr,
                     const float* __restrict__ Wi,
                     float* __restrict__ outR,
                     float* __restrict__ outI) {
    __shared__ float sXr[LINES_PER_WG * VDIM];   // 32 KB
    __shared__ float sXi[LINES_PER_WG * VDIM];   // 32 KB

    const int tid      = threadIdx.x;
    const int lineBase = blockIdx.x * LINES_PER_WG;

    // ---- Stage 64 lines (8192 contiguous floats per component) ----
    const float4* gR = (const float4*)(inR + (size_t)lineBase * VDIM);
    const float4* gI = HAS_IMAG ? (const float4*)(inI + (size_t)lineBase * VDIM)
                                : (const float4*)nullptr;
    for (int i = tid; i < (LINES_PER_WG * VDIM) / 4; i += 256) { // 2048 f4
        int flat = i * 4;
        int l = flat >> 7;                 // line within workgroup (0..63)
        int n = flat & 127;                // sample index (16B chunk)
        int p = lds_addr(l, n);            // contiguous 4-float run
        stage16(&sXr[p], gR + i);
        if (HAS_IMAG)
            stage16(&sXi[p], gI + i);
    }
    stage_fence();        // this wave's async copies complete
    __syncthreads();      // all waves' tiles visible

    // ---- WMMA GEMM: C(16x16 x4 tiles) = W(16x128) * X(128x16), complex ----
    const int wave = tid >> 5;          // 0..7 -> k tile
    const int lane = tid & 31;
    const int col  = lane & 15;         // A: row M ; B: column (line)
    const int hi   = lane >> 4;         // K-half select
    const int k0   = wave * 16;
    const int swz  = col << 2;          // LDS XOR swizzle for this lane

    const float* wrRow = Wr + (size_t)(k0 + col) * VDIM + hi * 2;
    const float* wiRow = Wi + (size_t)(k0 + col) * VDIM + hi * 2;

    v8f acc_rr[TILES];   // Wr*Xr
    v8f acc_ii[TILES];   // Wi*Xi
    v8f acc_ri[TILES];   // Wr*Xi
    v8f acc_ir[TILES];   // Wi*Xr
    #pragma unroll
    for (int t = 0; t < TILES; ++t) {
        acc_rr[t] = (v8f)0.0f;
        acc_ii[t] = (v8f)0.0f;
        acc_ri[t] = (v8f)0.0f;
        acc_ir[t] = (v8f)0.0f;
    }

    for (int kk = 0; kk < VDIM; kk += 4) {
        v2f aR = *(const v2f*)(wrRow + kk);   // 8B-aligned (even float idx)
        v2f aI = *(const v2f*)(wiRow + kk);
        const int e = (kk + hi * 2) ^ swz;    // even ^ mult-of-4 -> even

        #pragma unroll
        for (int t = 0; t < TILES; ++t) {
            const int rt = (t * 16 + col) * VDIM + e;
            v2f bR = *(const v2f*)&sXr[rt];   // conflict-free ds_load_b64

            acc_rr[t] = wmma_f32(aR, bR, acc_rr[t]);
            if (WRITE_IMAG)
                acc_ir[t] = wmma_f32(aI, bR, acc_ir[t]);
            if (HAS_IMAG) {
                v2f bI = *(const v2f*)&sXi[rt];
                acc_ii[t] = wmma_f32(aI, bI, acc_ii[t]);
                if (WRITE_IMAG)
                    acc_ri[t] = wmma_f32(aR, bI, acc_ri[t]);
            }
        }
    }

    // ---- Combine complex parts (compile-time selected) and store ----
    #pragma unroll
    for (int t = 0; t < TILES; ++t) {
        const size_t lcol = (size_t)(lineBase + t * 16 + col);
        #pragma unroll
        for (int v = 0; v < 8; ++v) {
            int k = k0 + v + hi * 8;                  // C/D VGPR layout
            size_t o = (size_t)k * NLINE + lcol;
            float cr = HAS_IMAG ? (acc_rr[t][v] - acc_ii[t][v])
                                : acc_rr[t][v];
            outR[o] = cr;
            if (WRITE_IMAG) {
                float ci = HAS_IMAG ? (acc_ri[t][v] + acc_ir[t][v])
                                    : acc_ir[t][v];
                outI[o] = ci;
            }
        }
    }
}

// ---------------------------------------------------------------------
// P = F1 * conj(F2), elementwise (float4-vectorized).
// ---------------------------------------------------------------------
__global__ __launch_bounds__(256)
void pointwise_conj_mul_kernel(const float4* __restrict__ ar,
                               const float4* __restrict__ ai,
                               const float4* __restrict__ br,
                               const float4* __restrict__ bi,
                               float4* __restrict__ pr,
                               float4* __restrict__ pi) {
    int i = blockIdx.x * 256 + threadIdx.x;
    float4 x1r = ar[i], x1i = ai[i], x2r = br[i], x2i = bi[i];
    float4 orr, oii;
    orr.x = x1r.x * x2r.x + x1i.x * x2i.x;
    orr.y = x1r.y * x2r.y + x1i.y * x2i.y;
    orr.z = x1r.z * x2r.z + x1i.z * x2i.z;
    orr.w = x1r.w * x2r.w + x1i.w * x2i.w;
    oii.x = x1i.x * x2r.x - x1r.x * x2i.x;
    oii.y = x1i.y * x2r.y - x1r.y * x2i.y;
    oii.z = x1i.z * x2r.z - x1r.z * x2i.z;
    oii.w = x1i.w * x2r.w - x1r.w * x2i.w;
    pr[i] = orr;
    pi[i] = oii;
}

// ---------------------------------------------------------------------
// Host-side orchestration (graph-capture safe: stream-only, no sync).
// Scratch layout (floats): 4x16384 twiddles, then 6 volume buffers
// (3 complex ping-pong volumes)  => ~50.6 MB required in d_ws.
// ---------------------------------------------------------------------
extern "C" void kernel_launch(void* const* d_in, const int* in_sizes, int n_in,
                              void* d_out, int out_size, void* d_ws, size_t ws_size,
                              hipStream_t stream) {
    const float* v1  = (const float*)d_in[0];   // [8,128,128,128] f32
    const float* v2  = (const float*)d_in[1];   // [8,128,128,128] f32
    float*       out = (float*)d_out;           // [8,128,128,128] f32
    float*       ws  = (float*)d_ws;

    float* wf_r = ws;                 // forward DFT matrix, real
    float* wf_i = wf_r + NLINE;       // forward DFT matrix, imag
    float* wv_r = wf_i + NLINE;       // inverse DFT matrix (scaled), real
    float* wv_i = wv_r + NLINE;       // inverse DFT matrix (scaled), imag
    float* b0r  = wv_i + NLINE;
    float* b0i  = b0r + VOL;
    float* b1r  = b0i + VOL;
    float* b1i  = b1r + VOL;
    float* b2r  = b1i + VOL;
    float* b2i  = b2r + VOL;

    twiddle_init_kernel<<<NLINE / 256, 256, 0, stream>>>(wf_r, wf_i, wv_r, wv_i);

    const dim3 gPass(NLINE / LINES_PER_WG);  // 256 workgroups per pass
    const dim3 gPt(VOL / (256 * 4));         // 2048 workgroups pointwise
    const int  B = 8;

    for (int b = 0; b < B; ++b) {
        const float* in1  = v1 + (size_t)b * VOL;
        const float* in2  = v2 + (size_t)b * VOL;
        float*       outb = out + (size_t)b * VOL;

        // F1 = FFT3(v1): real input -> 3 transposing passes -> b0
        dft_pass_kernel<false, true><<<gPass, 256, 0, stream>>>(in1, nullptr, wf_r, wf_i, b0r, b0i);
        dft_pass_kernel<true,  true><<<gPass, 256, 0, stream>>>(b0r, b0i,     wf_r, wf_i, b1r, b1i);
        dft_pass_kernel<true,  true><<<gPass, 256, 0, stream>>>(b1r, b1i,     wf_r, wf_i, b0r, b0i);

        // F2 = FFT3(v2) -> b1
        dft_pass_kernel<false, true><<<gPass, 256, 0, stream>>>(in2, nullptr, wf_r, wf_i, b1r, b1i);
        dft_pass_kernel<true,  true><<<gPass, 256, 0, stream>>>(b1r, b1i,     wf_r, wf_i, b2r, b2i);
        dft_pass_kernel<true,  true><<<gPass, 256, 0, stream>>>(b2r, b2i,     wf_r, wf_i, b1r, b1i);

        // P = F1 * conj(F2) -> b2
        pointwise_conj_mul_kernel<<<gPt, 256, 0, stream>>>(
            (const float4*)b0r, (const float4*)b0i,
            (const float4*)b1r, (const float4*)b1i,
            (float4*)b2r, (float4*)b2i);

        // out = IFFT3(P): 3 passes with scaled conjugate matrix; last pass
        // writes only the real part straight into d_out.
        dft_pass_kernel<true, true ><<<gPass, 256, 0, stream>>>(b2r, b2i, wv_r, wv_i, b0r, b0i);
        dft_pass_kernel<true, true ><<<gPass, 256, 0, stream>>>(b0r, b0i, wv_r, wv_i, b1r, b1i);
        dft_pass_kernel<true, false><<<gPass, 256, 0, stream>>>(b1r, b1i, wv_r, wv_i, outb, nullptr);
    }
}